// TrackingHead_28132035789033
// MI455X (gfx1250) — compile-verified
//
#include <hip/hip_runtime.h>
#include <hip/hip_bf16.h>
#include <math.h>

typedef float v2f __attribute__((ext_vector_type(2)));
typedef float v4f __attribute__((ext_vector_type(4)));
typedef float v8f __attribute__((ext_vector_type(8)));

#define WARP 32

__device__ __forceinline__ float rfl_f(float x) {
    return __uint_as_float(__builtin_amdgcn_readfirstlane(__float_as_uint(x)));
}

// ---------------------------------------------------------------------------
// Kernel 0: pad + transpose w1 [27,32] -> w1t [32,28] (col 27 zeroed) so the
// WMMA B-operand loads are unconditional aligned b64s.
// ---------------------------------------------------------------------------
__global__ void prep_w1_kernel(const float* __restrict__ w1,
                               float* __restrict__ w1t) {
    int i = (int)(blockIdx.x * blockDim.x + threadIdx.x);
    if (i < 32 * 28) {
        int n = i / 28, k = i % 28;
        w1t[i] = (k < 27) ? w1[k * 32 + n] : 0.f;
    }
}

// ---------------------------------------------------------------------------
// Kernel 1: one wave32 per query. All query-derived state is wave-uniform
// (readfirstlane -> SGPRs). Gather the 6x6 lattice patch per channel with
// clamped unconditional loads + uniform mask multiply, build 25 bilinear
// window responses in registers, accumulate dot(q,w) and ||w||^2,
// butterfly-reduce, softmax -> delta/score, emit vis_input[28].
// ---------------------------------------------------------------------------
__global__ __launch_bounds__(256) void corr_softmax_kernel(
    const float* __restrict__ qf,     // [B,Q,C]
    const float* __restrict__ fmap,   // [B,C,H,W]
    const float* __restrict__ pos,    // [B,Q,2]
    float* __restrict__ out_delta,    // [BQ,2]
    float* __restrict__ out_score,    // [BQ]
    float* __restrict__ vis,          // [BQ,28]
    int BQ)
{
    const int H = 128, W = 128, C = 256;
    int wave = __builtin_amdgcn_readfirstlane(
        (int)((blockIdx.x * blockDim.x + threadIdx.x) >> 5));
    if (wave >= BQ) return;
    int lane = (int)(threadIdx.x & 31);
    int b = wave >> 12;                     // Q = 4096

    float px = rfl_f(pos[wave * 2 + 0]);
    float py = rfl_f(pos[wave * 2 + 1]);
    float fx = (px + 0.5f) * 0.25f - 0.5f;
    float fy = (py + 0.5f) * 0.25f - 0.5f;
    float x0f = floorf(fx), y0f = floorf(fy);
    float wx1 = fx - x0f, wx0 = 1.f - wx1;
    float wy1 = fy - y0f, wy0 = 1.f - wy1;
    int ix0 = (int)x0f, iy0 = (int)y0f;
    float w00 = wy0 * wx0, w01 = wy0 * wx1;
    float w10 = wy1 * wx0, w11 = wy1 * wx1;

    // uniform lattice offsets (clamped) + uniform 0/1 masks
    int   off6[6][6];
    float m6[6][6];
#pragma unroll
    for (int jy = 0; jy < 6; ++jy) {
        int yy = iy0 + jy - 2;
        bool yv = (yy >= 0) && (yy < H);
        int yc = yv ? yy : (yy < 0 ? 0 : H - 1);
#pragma unroll
        for (int jx = 0; jx < 6; ++jx) {
            int xx = ix0 + jx - 2;
            bool xv = (xx >= 0) && (xx < W);
            int xc = xv ? xx : (xx < 0 ? 0 : W - 1);
            off6[jy][jx] = yc * W + xc;
            m6[jy][jx]   = (xv && yv) ? 1.f : 0.f;
        }
    }

    float dotp[25], nrm[25];
#pragma unroll
    for (int p = 0; p < 25; ++p) { dotp[p] = 0.f; nrm[p] = 0.f; }
    float qn = 0.f;

    const float* qbase = qf + (size_t)wave * C;
    const float* fb    = fmap + (size_t)b * C * (H * W);

#pragma unroll 1
    for (int j = 0; j < 8; ++j) {
        int c = lane + 32 * j;
        float qc = qbase[c];
        qn = fmaf(qc, qc, qn);
        const float* fc = fb + (size_t)c * (H * W);

        float g[6][6];
#pragma unroll
        for (int jy = 0; jy < 6; ++jy) {
#pragma unroll
            for (int jx = 0; jx < 6; ++jx) {
                g[jy][jx] = fc[off6[jy][jx]] * m6[jy][jx];
            }
        }
#pragma unroll
        for (int iy = 0; iy < 5; ++iy) {
#pragma unroll
            for (int ix = 0; ix < 5; ++ix) {
                float wv = w00 * g[iy][ix]     + w01 * g[iy][ix + 1]
                         + w10 * g[iy + 1][ix] + w11 * g[iy + 1][ix + 1];
                int p = iy * 5 + ix;
                dotp[p] = fmaf(qc, wv, dotp[p]);
                nrm[p]  = fmaf(wv, wv, nrm[p]);
            }
        }
    }

    // cross-lane (channel) reduction: butterfly; every lane gets full sums
#pragma unroll
    for (int p = 0; p < 25; ++p) {
        float d = dotp[p], n = nrm[p];
#pragma unroll
        for (int m = 16; m >= 1; m >>= 1) {
            d += __shfl_xor(d, m, WARP);
            n += __shfl_xor(n, m, WARP);
        }
        dotp[p] = d; nrm[p] = n;
    }
#pragma unroll
    for (int m = 16; m >= 1; m >>= 1) qn += __shfl_xor(qn, m, WARP);

    // logits = cos-sim / TEMP (TEMP = 0.1)
    float qinv = 1.f / fmaxf(sqrtf(qn), 1e-12f);
    float lg[25];
    float mx = -1e30f;
#pragma unroll
    for (int p = 0; p < 25; ++p) {
        float wn = fmaxf(sqrtf(nrm[p]), 1e-12f);
        lg[p] = (dotp[p] * qinv / wn) * 10.0f;
        mx = fmaxf(mx, lg[p]);
    }
    float se = 0.f;
#pragma unroll
    for (int p = 0; p < 25; ++p) { lg[p] = __expf(lg[p] - mx); se += lg[p]; }
    float inv = 1.f / se;

    float dx = 0.f, dy = 0.f, peak = 0.f, ent = 0.f;
#pragma unroll
    for (int iy = 0; iy < 5; ++iy) {
#pragma unroll
        for (int ix = 0; ix < 5; ++ix) {
            int p = iy * 5 + ix;
            float pp = lg[p] * inv;
            lg[p] = pp;
            dx = fmaf(pp, (float)(ix - 2) * 4.0f, dx);
            dy = fmaf(pp, (float)(iy - 2) * 4.0f, dy);
            peak = fmaxf(peak, pp);
            ent -= pp * __logf(pp + 1e-8f);
        }
    }
    const float maxent = 3.2188758248682006f;  // log(25)
    float conf  = fminf(fmaxf(1.f - ent / maxent, 0.f), 1.f);
    float score = fminf(fmaxf(0.5f * peak + 0.5f * conf, 0.f), 1.f);
    float dmag  = sqrtf(dx * dx + dy * dy);

    if (lane == 0) {
        v2f d2; d2.x = dx; d2.y = dy;
        *(v2f*)(out_delta + (size_t)wave * 2) = d2;
        out_score[wave] = score;
        float* v = vis + (size_t)wave * 28;    // 112B row stride -> 16B aligned
        v4f q0 = { lg[0],  lg[1],  lg[2],  lg[3]  };
        v4f q1 = { lg[4],  lg[5],  lg[6],  lg[7]  };
        v4f q2 = { lg[8],  lg[9],  lg[10], lg[11] };
        v4f q3 = { lg[12], lg[13], lg[14], lg[15] };
        v4f q4 = { lg[16], lg[17], lg[18], lg[19] };
        v4f q5 = { lg[20], lg[21], lg[22], lg[23] };
        v4f q6 = { lg[24], peak,   dmag,   0.f    };
        *(v4f*)(v +  0) = q0;  *(v4f*)(v +  4) = q1;
        *(v4f*)(v +  8) = q2;  *(v4f*)(v + 12) = q3;
        *(v4f*)(v + 16) = q4;  *(v4f*)(v + 20) = q5;
        *(v4f*)(v + 24) = q6;
    }
}

// ---------------------------------------------------------------------------
// Kernel 2: visibility MLP with fp32 WMMA.
// [BQ x 28] @ [28 x 32] via V_WMMA_F32_16X16X4_F32 (7 k-steps, 2 N-tiles),
// exact GELU, then 32->1 via per-lane scaling + 16-lane shuffle reduction.
// One wave per 16-row M-tile; all operand loads unconditional & vectorized.
// ---------------------------------------------------------------------------
__global__ __launch_bounds__(256) void vis_mlp_wmma_kernel(
    const float* __restrict__ vis,   // [BQ,28] (col 27 zero)
    const float* __restrict__ w1t,   // [32,28] transposed+padded
    const float* __restrict__ b1,    // [32]
    const float* __restrict__ w2,    // [32,1]
    const float* __restrict__ b2,    // [1]
    float* __restrict__ out_logit,   // [BQ]
    int BQ)
{
    int lane  = (int)(threadIdx.x & 31);
    int tile  = __builtin_amdgcn_readfirstlane(
        (int)((blockIdx.x * blockDim.x + threadIdx.x) >> 5));
    int ntile = (BQ + 15) >> 4;
    if (tile >= ntile) return;

    int row0  = tile * 16;
    int mrow  = lane & 15;          // A: M row held by this lane
    int khalf = (lane >> 4) * 2;    // A/B: K sub-offset (0 or 2)
    int ncol  = lane & 15;          // B/C/D: N column

    // init accumulators with bias (bias depends only on N)
    float bias0 = b1[ncol];
    float bias1 = b1[16 + ncol];
    v8f c0, c1;
#pragma unroll
    for (int r = 0; r < 8; ++r) { c0[r] = bias0; c1[r] = bias1; }

    int arow = row0 + mrow;
    if (arow >= BQ) arow = BQ - 1;               // clamp: load safe, store guarded
    const float* avp = vis + (size_t)arow * 28;
    const float* bp0 = w1t + (size_t)ncol * 28;
    const float* bp1 = w1t + (size_t)(16 + ncol) * 28;

#pragma unroll
    for (int t = 0; t < 7; ++t) {
        int k = 4 * t + khalf;
        v2f a   = *(const v2f*)(avp + k);        // 8B-aligned
        v2f bm0 = *(const v2f*)(bp0 + k);
        v2f bm1 = *(const v2f*)(bp1 + k);
        c0 = __builtin_amdgcn_wmma_f32_16x16x4_f32(
                 false, a, false, bm0, (short)0, c0, false, false);
        c1 = __builtin_amdgcn_wmma_f32_16x16x4_f32(
                 false, a, false, bm1, (short)0, c1, false, false);
    }

    // exact GELU: 0.5*x*(1+erf(x/sqrt(2)))
#pragma unroll
    for (int r = 0; r < 8; ++r) {
        float x0 = c0[r];
        float x1 = c1[r];
        c0[r] = 0.5f * x0 * (1.f + erff(x0 * 0.70710678118654752f));
        c1[r] = 0.5f * x1 * (1.f + erff(x1 * 0.70710678118654752f));
    }

    // layer 2: h[16x32] @ w2[32]; D layout: VGPR r, lane l -> row r + 8*(l/16)
    float w2a = w2[ncol];
    float w2b = w2[16 + ncol];
    float acc[8];
#pragma unroll
    for (int r = 0; r < 8; ++r) acc[r] = c0[r] * w2a + c1[r] * w2b;
#pragma unroll
    for (int r = 0; r < 8; ++r) {
#pragma unroll
        for (int m = 8; m >= 1; m >>= 1) acc[r] += __shfl_xor(acc[r], m, WARP);
    }
    float bb = b2[0];
    if (ncol == 0) {                 // lane 0 -> rows 0..7, lane 16 -> rows 8..15
        int mbase = row0 + ((lane >> 4) << 3);
#pragma unroll
        for (int r = 0; r < 8; ++r) {
            int row = mbase + r;
            if (row < BQ) out_logit[row] = acc[r] + bb;
        }
    }
}

extern "C" void kernel_launch(void* const* d_in, const int* in_sizes, int n_in,
                              void* d_out, int out_size, void* d_ws, size_t ws_size,
                              hipStream_t stream) {
    const float* qf   = (const float*)d_in[0];  // [B,Q,C]
    const float* fmap = (const float*)d_in[1];  // [B,C,H,W]
    const float* pos  = (const float*)d_in[2];  // [B,Q,2]
    const float* w1   = (const float*)d_in[3];  // [27,32]
    const float* b1   = (const float*)d_in[4];  // [32]
    const float* w2   = (const float*)d_in[5];  // [32,1]
    const float* b2   = (const float*)d_in[6];  // [1]

    int BQ = in_sizes[2] / 2;                   // B*Q = 16384

    float* out    = (float*)d_out;
    float* delta  = out;                        // [BQ*2]
    float* vlogit = out + (size_t)BQ * 2;       // [BQ]
    float* score  = out + (size_t)BQ * 3;       // [BQ]

    float* vis = (float*)d_ws;                  // [BQ*28]
    float* w1t = vis + (size_t)BQ * 28;         // [32*28]

    const int threads = 256;
    const int waves_per_block = threads / 32;

    prep_w1_kernel<<<4, threads, 0, stream>>>(w1, w1t);

    int g1 = (BQ + waves_per_block - 1) / waves_per_block;   // 1 wave / query
    corr_softmax_kernel<<<g1, threads, 0, stream>>>(
        qf, fmap, pos, delta, score, vis, BQ);

    int ntile = (BQ + 15) / 16;                              // 1 wave / 16 rows
    int g2 = (ntile + waves_per_block - 1) / waves_per_block;
    vis_mlp_wmma_kernel<<<g2, threads, 0, stream>>>(
        vis, w1t, b1, w2, b2, vlogit, BQ);
}